// Memory_6932077216211
// MI455X (gfx1250) — compile-verified
//
#include <hip/hip_runtime.h>
#include <cstdint>
#include <cstddef>

// Problem constants (match reference)
#define B_    1024
#define N_    512
#define M_    256
#define OUTS  (N_ + M_ + N_*M_)   // 131840 floats per batch
#define EPSA  1e-16f
#define EPSC  1e-8f
#define TS    68                  // LDS tile row stride in floats (272B: 16B aligned, conflict-mitigating)

typedef float v2f __attribute__((ext_vector_type(2)));
typedef float v8f __attribute__((ext_vector_type(8)));

#if __has_builtin(__builtin_amdgcn_wmma_f32_16x16x4_f32)
#define HAVE_WMMA4 1
#else
#define HAVE_WMMA4 0
#endif

#if __has_builtin(__builtin_amdgcn_global_load_async_to_lds_b128)
#define HAVE_ASYNC 1
// Builtin signature (from clang diagnostic): arg0 = v4i in addrspace(1), arg1 = v4i in addrspace(3).
typedef int v4i_ __attribute__((__vector_size__(16)));
typedef __attribute__((address_space(1))) v4i_ as1_v4i;
typedef __attribute__((address_space(3))) v4i_ as3_v4i;
#else
#define HAVE_ASYNC 0
#endif

__device__ __forceinline__ void wait_async_0() {
#if __has_builtin(__builtin_amdgcn_s_wait_asynccnt)
  __builtin_amdgcn_s_wait_asynccnt(0);
#else
  asm volatile("s_wait_asynccnt 0" ::: "memory");
#endif
}

__device__ __forceinline__ float blockSum256(float v, float* sred) {
  const int t = threadIdx.x;
  sred[t] = v;
  __syncthreads();
  #pragma unroll
  for (int off = 128; off > 0; off >>= 1) {
    if (t < off) sred[t] += sred[t + off];
    __syncthreads();
  }
  const float r = sred[0];
  __syncthreads();
  return r;
}

__device__ __forceinline__ float blockMax256(float v, float* sred) {
  const int t = threadIdx.x;
  sred[t] = v;
  __syncthreads();
  #pragma unroll
  for (int off = 128; off > 0; off >>= 1) {
    if (t < off) sred[t] = fmaxf(sred[t], sred[t + off]);
    __syncthreads();
  }
  const float r = sred[0];
  __syncthreads();
  return r;
}

__global__ __launch_bounds__(256) void ntm_step_kernel(
    const float* __restrict__ mem, const float* __restrict__ key,
    const float* __restrict__ beta, const float* __restrict__ g,
    const float* __restrict__ s, const float* __restrict__ gamma,
    const float* __restrict__ w_prev, const float* __restrict__ e,
    const float* __restrict__ a, float* __restrict__ out)
{
  __shared__ float lkey[M_];          // key + eps
  __shared__ float lnum[N_];
  __shared__ float lssq[N_];
  __shared__ float lwg[N_];
  __shared__ float lw[N_];
  __shared__ float sred[256];
  __shared__ float ltile[8 * 16 * TS]; // per-wave 16x64 staging tiles (reused as r-reduction buf)

  const int t    = threadIdx.x;
  const int lane = t & 31;
  const int wv   = t >> 5;
  const int b    = blockIdx.x;

  const float betav  = beta[b];
  const float gv     = g[b];
  const float gammav = gamma[b];
  const float s0 = s[b*3+0], s1 = s[b*3+1], s2 = s[b*3+2];

  // ---- key (+eps) into LDS; key norm (clamped) ----
  const float kv = key[(size_t)b * M_ + t] + EPSA;
  lkey[t] = kv;                              // visible after first barrier inside blockSum256
  const float kss   = blockSum256(kv * kv, sred);
  const float knorm = fmaxf(sqrtf(kss), EPSC);

  // ================= Phase 1: per-row dot(mem+eps, key+eps) and ||mem+eps||^2 =================
  float* tile = &ltile[wv * 16 * TS];
  const size_t membase = (size_t)b * (size_t)(N_ * M_);
  const int rr = lane & 15;   // row within 16-row subgroup
  const int h  = lane >> 4;   // half-wave selector

  for (int sub = 0; sub < 4; ++sub) {
    const int rowbase = wv * 64 + sub * 16;
    v8f  acc = {0.f, 0.f, 0.f, 0.f, 0.f, 0.f, 0.f, 0.f};
    float numacc = 0.f;
    float ssq    = 0.f;

    for (int mc = 0; mc < 4; ++mc) {
      const int m0 = mc * 64;
      // ---- stage 16 rows x 64 cols into LDS (coalesced b128) ----
      #pragma unroll
      for (int it = 0; it < 8; ++it) {
        const int elem = it * 128 + lane * 4;   // float index into 16x64 tile
        const int r = elem >> 6;
        const int c = elem & 63;
        const float* gp = mem + membase + (size_t)(rowbase + r) * M_ + (m0 + c);
        float* lp = tile + r * TS + c;
#if HAVE_ASYNC
        __builtin_amdgcn_global_load_async_to_lds_b128(
            (as1_v4i*)(uintptr_t)gp,
            (as3_v4i*)(unsigned)(uintptr_t)lp, 0, 0);
#else
        const float4 v4 = *(const float4*)gp;
        lp[0] = v4.x; lp[1] = v4.y; lp[2] = v4.z; lp[3] = v4.w;
#endif
      }
#if HAVE_ASYNC
      wait_async_0();
#endif
      // ---- ssq partial: lane handles row rr, interleaved columns 2*c2+h (bank-conflict-free) ----
      #pragma unroll
      for (int c2 = 0; c2 < 32; ++c2) {
        const float v = tile[rr * TS + 2 * c2 + h] + EPSA;
        ssq = fmaf(v, v, ssq);
      }
#if HAVE_WMMA4
      // ---- num via V_WMMA_F32_16X16X4_F32: A = 16 mem rows x 4 cols, B = key chunk
      //      broadcast across all 16 columns -> D[i,j] = partial num[row i] (any j) ----
      #pragma unroll
      for (int k = 0; k < 16; ++k) {
        const int mk = k * 4 + h * 2;  // lanes<16 feed K0,K1; lanes>=16 feed K2,K3 (ISA 16x4 A layout)
        v2f av, bv;
        av.x = tile[rr * TS + mk]     + EPSA;
        av.y = tile[rr * TS + mk + 1] + EPSA;
        bv.x = lkey[m0 + mk];
        bv.y = lkey[m0 + mk + 1];
        acc = __builtin_amdgcn_wmma_f32_16x16x4_f32(false, av, false, bv,
                                                    (short)0, acc, false, false);
      }
#else
      #pragma unroll
      for (int c2 = 0; c2 < 32; ++c2) {
        const int c = 2 * c2 + h;
        const float v = tile[rr * TS + c] + EPSA;
        numacc = fmaf(v, lkey[m0 + c], numacc);
      }
#endif
    }
    // ---- write ssq (combine half-waves) ----
    {
      const float other = __shfl_xor(ssq, 16, 32);
      const float tot = ssq + other;
      if (lane < 16) lssq[rowbase + lane] = tot;
    }
#if HAVE_WMMA4
    // D col 0 lives in lanes 0 (M=0..7) and 16 (M=8..15)
    if ((lane & 15) == 0) {
      const int off = h * 8;
      #pragma unroll
      for (int i = 0; i < 8; ++i) lnum[rowbase + off + i] = acc[i];
    }
    (void)numacc;
#else
    {
      const float other = __shfl_xor(numacc, 16, 32);
      if (lane < 16) lnum[rowbase + lane] = numacc + other;
    }
#endif
  }
  __syncthreads();

  // ================= softmax / gate / shift / sharpen (each thread owns n=t and n=t+256) =================
  const int n0 = t, n1 = t + 256;
  const float x0 = betav * (lnum[n0] / (fmaxf(sqrtf(lssq[n0]), EPSC) * knorm));
  const float x1 = betav * (lnum[n1] / (fmaxf(sqrtf(lssq[n1]), EPSC) * knorm));
  const float xmax = blockMax256(fmaxf(x0, x1), sred);
  const float e0 = expf(x0 - xmax);
  const float e1 = expf(x1 - xmax);
  const float esum = blockSum256(e0 + e1, sred);
  const float einv = 1.f / esum;
  lwg[n0] = gv * (e0 * einv) + (1.f - gv) * w_prev[(size_t)b * N_ + n0];
  lwg[n1] = gv * (e1 * einv) + (1.f - gv) * w_prev[(size_t)b * N_ + n1];
  __syncthreads();

  const float wt0 = lwg[(n0 + N_ - 1) & (N_ - 1)] * s0 + lwg[n0] * s1 + lwg[(n0 + 1) & (N_ - 1)] * s2;
  const float wt1 = lwg[(n1 + N_ - 1) & (N_ - 1)] * s0 + lwg[n1] * s1 + lwg[(n1 + 1) & (N_ - 1)] * s2;
  const float wp0 = powf(wt0, gammav);
  const float wp1 = powf(wt1, gammav);
  const float pinv = 1.f / (blockSum256(wp0 + wp1, sred) + EPSA);
  const float w0 = wp0 * pinv, w1 = wp1 * pinv;
  lw[n0] = w0; lw[n1] = w1;
  const size_t ob = (size_t)b * OUTS;
  out[ob + n0] = w0;
  out[ob + n1] = w1;
  __syncthreads();

  // ================= Phase 2: read r = w . mem, and new_mem = mem*(1 - w e) + w a =================
  // lane owns columns [4*lane..4*lane+3] and [128+4*lane..+3] (fully coalesced b128)
  const float4 ev0 = *(const float4*)(e + (size_t)b * M_ + 4 * lane);
  const float4 ev1 = *(const float4*)(e + (size_t)b * M_ + 128 + 4 * lane);
  const float4 av0 = *(const float4*)(a + (size_t)b * M_ + 4 * lane);
  const float4 av1 = *(const float4*)(a + (size_t)b * M_ + 128 + 4 * lane);
  float4 r0 = {0.f, 0.f, 0.f, 0.f}, r1 = {0.f, 0.f, 0.f, 0.f};
  float* nmo = out + ob + N_ + M_;

  for (int row = wv * 64; row < wv * 64 + 64; ++row) {
    const float wn = lw[row];
    const float* mrow = mem + membase + (size_t)row * M_;
    const float4 m0v = *(const float4*)(mrow + 4 * lane);
    const float4 m1v = *(const float4*)(mrow + 128 + 4 * lane);
    r0.x = fmaf(wn, m0v.x, r0.x); r0.y = fmaf(wn, m0v.y, r0.y);
    r0.z = fmaf(wn, m0v.z, r0.z); r0.w = fmaf(wn, m0v.w, r0.w);
    r1.x = fmaf(wn, m1v.x, r1.x); r1.y = fmaf(wn, m1v.y, r1.y);
    r1.z = fmaf(wn, m1v.z, r1.z); r1.w = fmaf(wn, m1v.w, r1.w);
    float4 o0, o1;
    o0.x = m0v.x * (1.f - wn * ev0.x) + wn * av0.x;
    o0.y = m0v.y * (1.f - wn * ev0.y) + wn * av0.y;
    o0.z = m0v.z * (1.f - wn * ev0.z) + wn * av0.z;
    o0.w = m0v.w * (1.f - wn * ev0.w) + wn * av0.w;
    o1.x = m1v.x * (1.f - wn * ev1.x) + wn * av1.x;
    o1.y = m1v.y * (1.f - wn * ev1.y) + wn * av1.y;
    o1.z = m1v.z * (1.f - wn * ev1.z) + wn * av1.z;
    o1.w = m1v.w * (1.f - wn * ev1.w) + wn * av1.w;
    *(float4*)(nmo + (size_t)row * M_ + 4 * lane)       = o0;
    *(float4*)(nmo + (size_t)row * M_ + 128 + 4 * lane) = o1;
  }

  // reduce partial r across the 8 waves (reuse ltile; phase-1 use is long done)
  float* rbuf = ltile;
  {
    const int base = wv * 256;
    rbuf[base + 4 * lane + 0] = r0.x; rbuf[base + 4 * lane + 1] = r0.y;
    rbuf[base + 4 * lane + 2] = r0.z; rbuf[base + 4 * lane + 3] = r0.w;
    rbuf[base + 128 + 4 * lane + 0] = r1.x; rbuf[base + 128 + 4 * lane + 1] = r1.y;
    rbuf[base + 128 + 4 * lane + 2] = r1.z; rbuf[base + 128 + 4 * lane + 3] = r1.w;
  }
  __syncthreads();
  float rs = 0.f;
  #pragma unroll
  for (int w8 = 0; w8 < 8; ++w8) rs += rbuf[w8 * 256 + t];
  out[ob + N_ + t] = rs;
}

extern "C" void kernel_launch(void* const* d_in, const int* in_sizes, int n_in,
                              void* d_out, int out_size, void* d_ws, size_t ws_size,
                              hipStream_t stream) {
  (void)in_sizes; (void)n_in; (void)out_size; (void)d_ws; (void)ws_size;
  const float* mem    = (const float*)d_in[0];
  const float* key    = (const float*)d_in[1];
  const float* beta   = (const float*)d_in[2];
  const float* g      = (const float*)d_in[3];
  const float* s      = (const float*)d_in[4];
  const float* gamma  = (const float*)d_in[5];
  const float* w_prev = (const float*)d_in[6];
  const float* e      = (const float*)d_in[7];
  const float* a      = (const float*)d_in[8];
  float* out = (float*)d_out;
  ntm_step_kernel<<<B_, 256, 0, stream>>>(mem, key, beta, g, s, gamma, w_prev, e, a, out);
}